// HAAML_51960514347637
// MI455X (gfx1250) — compile-verified
//
#include <hip/hip_runtime.h>
#include <math.h>

typedef float v2f __attribute__((ext_vector_type(2)));
typedef float v8f __attribute__((ext_vector_type(8)));

#define BB   4096
#define DD   128
#define CC   50000
#define SS   64.0f
#define M0C  0.5f
#define TT   1.2f
#define LW   10.0f
#define COSM 0.8775825618903728f
#define SINM 0.479425538604203f
#define PIF  3.14159265358979323846f

#define LDSK 132           // 128 + 4 pad (bank-conflict avoidance, 528B row = 16B aligned)

// ---------------------------------------------------------------------------
// Kernel 1: per-row inverse norms for x (rows 0..B-1) and weight (rows B..B+C-1)
// renorm(2,0,1e-5).mul(1e5):  inv = (n > 1e-5) ? 1/n : 1e5
// ---------------------------------------------------------------------------
__global__ __launch_bounds__(128)
void haaml_rownorm(const float* __restrict__ x, const float* __restrict__ w,
                   float* __restrict__ xinv, float* __restrict__ winv)
{
    int row = blockIdx.x;
    int t = threadIdx.x;
    const float* src;
    float* dst;
    if (row < BB) { src = x + (size_t)row * DD;         dst = xinv + row; }
    else          { src = w + (size_t)(row - BB) * DD;  dst = winv + (row - BB); }
    float v = src[t];
    float s = v * v;
    #pragma unroll
    for (int off = 16; off > 0; off >>= 1) s += __shfl_xor(s, off, 32);
    __shared__ float red[4];
    if ((t & 31) == 0) red[t >> 5] = s;
    __syncthreads();
    if (t == 0) {
        float n = sqrtf(red[0] + red[1] + red[2] + red[3]);
        *dst = (n > 1e-5f) ? (1.0f / n) : 1e5f;
    }
}

// ---------------------------------------------------------------------------
// Kernel 2: gt = clip(<xn_i, wn_label>, -1, 1); ctm = gt*cos(m0) - sin*sin(m0)
// Also zeroes the hard-stat accumulators (graph-replay safe re-init).
// ---------------------------------------------------------------------------
__global__ __launch_bounds__(128)
void haaml_gt(const float* __restrict__ x, const float* __restrict__ w,
              const float* __restrict__ xinv, const float* __restrict__ winv,
              const int* __restrict__ label,
              float* __restrict__ gtv, float* __restrict__ ctmv,
              float* __restrict__ hsum, float* __restrict__ hcnt)
{
    int i = blockIdx.x;
    int t = threadIdx.x;
    int lab = label[i];
    float s = x[(size_t)i * DD + t] * w[(size_t)lab * DD + t];
    #pragma unroll
    for (int off = 16; off > 0; off >>= 1) s += __shfl_xor(s, off, 32);
    __shared__ float red[4];
    if ((t & 31) == 0) red[t >> 5] = s;
    __syncthreads();
    if (t == 0) {
        float dot = (red[0] + red[1] + red[2] + red[3]) * xinv[i] * winv[lab];
        float g = fminf(1.0f, fmaxf(-1.0f, dot));
        float st = sqrtf(fmaxf(0.0f, 1.0f - g * g));
        gtv[i]  = g;
        ctmv[i] = g * COSM - st * SINM;
        hsum[i] = 0.0f;
        hcnt[i] = 0.0f;
    }
}

// ---------------------------------------------------------------------------
// Kernel 3: fused WMMA GEMM + clamp + *S NT-store + hard-example statistics.
// Workgroup: 256 threads = 8 waves. WG tile: 128 M x 80 N.
// Wave tile: 16 M x 80 N = 5 accumulator chains of v_wmma_f32_16x16x4_f32.
// Weight tile (80 rows x 128 k) staged once in LDS, pre-scaled by winv.
// A fragments (whole K) live in registers, pre-scaled by xinv.
// ---------------------------------------------------------------------------
__global__ __launch_bounds__(256)
void haaml_gemm(const float* __restrict__ x, const float* __restrict__ w,
                const float* __restrict__ xinv, const float* __restrict__ winv,
                const float* __restrict__ ctmv, const int* __restrict__ label,
                float* __restrict__ out,
                float* __restrict__ hsum, float* __restrict__ hcnt)
{
    __shared__ __align__(16) float LB[80 * LDSK];

    const int t    = threadIdx.x;
    const int lane = t & 31;
    const int wv   = t >> 5;         // wave 0..7
    const int l16  = lane & 15;
    const int h    = lane >> 4;      // half-wave select (K pair / M+8)
    const int nbase = blockIdx.x * 80;
    const int mbase = blockIdx.y * 128;

    // ---- cooperative stage of B tile (scaled by winv) : 80 x 128 f32 ----
    {
        int row = t >> 4;            // 0..15
        int kq  = (t & 15) * 8;      // 0..120
        #pragma unroll
        for (int rr = 0; rr < 5; ++rr) {
            int r = rr * 16 + row;   // 0..79
            float wsc = winv[nbase + r];
            const float4* src = (const float4*)(w + (size_t)(nbase + r) * DD + kq);
            float4 v0 = src[0];
            float4 v1 = src[1];
            float4 s0 = make_float4(v0.x * wsc, v0.y * wsc, v0.z * wsc, v0.w * wsc);
            float4 s1 = make_float4(v1.x * wsc, v1.y * wsc, v1.z * wsc, v1.w * wsc);
            float* dst = &LB[r * LDSK + kq];
            *(float4*)dst       = s0;
            *((float4*)dst + 1) = s1;
        }
    }

    // ---- per-wave A fragments: 16 rows x full K=128, in registers ----
    const int m0w  = mbase + wv * 16;
    const int mrow = m0w + l16;                 // lane's fixed A row
    const float xs = xinv[mrow];
    const float* arow = x + (size_t)mrow * DD + 2 * h;
    v2f a[32];
    #pragma unroll
    for (int kt = 0; kt < 32; ++kt) {
        v2f av = *(const v2f*)(arow + 4 * kt);  // 8B-aligned
        a[kt].x = av.x * xs;
        a[kt].y = av.y * xs;
    }

    // ---- per-lane row metadata for epilogue (8 C/D rows per lane) ----
    float ctm_r[8];
    int   lab_r[8];
    #pragma unroll
    for (int r = 0; r < 8; ++r) {
        int m = m0w + r + 8 * h;
        ctm_r[r] = ctmv[m];
        lab_r[r] = label[m];
    }

    __syncthreads();

    // ---- main WMMA loop: 5 independent accumulation chains ----
    v8f acc[5];
    #pragma unroll
    for (int j = 0; j < 5; ++j)
        #pragma unroll
        for (int e = 0; e < 8; ++e) acc[j][e] = 0.0f;

    const float* bbase = &LB[l16 * LDSK + 2 * h];
    #pragma unroll
    for (int kt = 0; kt < 32; ++kt) {
        #pragma unroll
        for (int j = 0; j < 5; ++j) {
            v2f b = *(const v2f*)(bbase + j * 16 * LDSK + 4 * kt);   // ds_load_b64
            acc[j] = __builtin_amdgcn_wmma_f32_16x16x4_f32(
                false, a[kt], false, b, (short)0, acc[j], false, false);
        }
    }

    // ---- epilogue: clamp, scale, NT-store, hard stats ----
    float hs[8], hc[8];
    #pragma unroll
    for (int r = 0; r < 8; ++r) { hs[r] = 0.0f; hc[r] = 0.0f; }

    #pragma unroll
    for (int j = 0; j < 5; ++j) {
        int n = nbase + j * 16 + l16;
        #pragma unroll
        for (int r = 0; r < 8; ++r) {
            float c = fminf(1.0f, fmaxf(-1.0f, acc[j][r]));
            int m = m0w + r + 8 * h;
            __builtin_nontemporal_store(c * SS, out + (size_t)m * CC + n);
            if ((c > ctm_r[r]) && (n != lab_r[r])) {
                hs[r] += c - ctm_r[r];
                hc[r] += 1.0f;
            }
        }
    }

    // reduce within each 16-lane group (rows r | r+8 live in half-waves)
    #pragma unroll
    for (int r = 0; r < 8; ++r) {
        #pragma unroll
        for (int off = 1; off < 16; off <<= 1) {
            hs[r] += __shfl_xor(hs[r], off, 16);
            hc[r] += __shfl_xor(hc[r], off, 16);
        }
    }
    if (l16 == 0) {
        #pragma unroll
        for (int r = 0; r < 8; ++r) {
            int m = m0w + r + 8 * h;
            atomicAdd(&hsum[m], hs[r]);
            atomicAdd(&hcnt[m], hc[r]);
        }
    }
}

// ---------------------------------------------------------------------------
// Kernel 4: per-row finalize: H, adaptive margin, patch label column, outputs
// ---------------------------------------------------------------------------
__global__ __launch_bounds__(256)
void haaml_finalize(const float* __restrict__ gtv,
                    const float* __restrict__ hsum, const float* __restrict__ hcnt,
                    const int* __restrict__ label, float* __restrict__ out)
{
    int i = blockIdx.x * blockDim.x + threadIdx.x;
    if (i >= BB) return;
    float hc = fminf(fmaxf(hcnt[i], 1.0f), (float)CC);
    float H  = hsum[i] / hc;
    float g  = gtv[i];
    float st = sqrtf(fmaxf(0.0f, 1.0f - g * g));
    float nm = M0C + TT * logf(H + 1.0f);
    if (nm > 0.75f) nm = 0.0f;
    float ctnm = g * cosf(nm) - st * sinf(nm);
    float thr  = cosf(PIF - nm);
    float mm   = sinf(PIF - nm) * nm;
    float ng   = (g > thr) ? ctnm : (g - mm);

    out[(size_t)i * CC + label[i]] = SS * ng;
    size_t base = (size_t)BB * CC + 1;
    out[base + i]          = g;    // gt  [B,1]
    out[base + BB + i]     = nm;   // new_m [1,B]
    out[base + 2 * BB + i] = H;    // H   [1,B]
}

// ---------------------------------------------------------------------------
// Kernel 5: hard_regularizer = LOSS_W * mean(H)   (single block, fixed order)
// ---------------------------------------------------------------------------
__global__ __launch_bounds__(1024)
void haaml_reg(const float* __restrict__ Hsrc, float* __restrict__ out_scalar)
{
    __shared__ float red[1024];
    int t = threadIdx.x;
    float s = Hsrc[t] + Hsrc[t + 1024] + Hsrc[t + 2048] + Hsrc[t + 3072];
    red[t] = s;
    __syncthreads();
    for (int off = 512; off > 0; off >>= 1) {
        if (t < off) red[t] += red[t + off];
        __syncthreads();
    }
    if (t == 0) *out_scalar = LW * red[0] / (float)BB;
}

// ---------------------------------------------------------------------------
extern "C" void kernel_launch(void* const* d_in, const int* in_sizes, int n_in,
                              void* d_out, int out_size, void* d_ws, size_t ws_size,
                              hipStream_t stream)
{
    const float* x     = (const float*)d_in[0];   // [B, D]
    const float* w     = (const float*)d_in[1];   // [C, D]
    const int*   label = (const int*)d_in[2];     // [B]
    float*       out   = (float*)d_out;           // B*C + 1 + 3*B floats

    // workspace layout (floats): xinv[B] | winv[C] | gt[B] | ctm[B] | hsum[B] | hcnt[B]
    float* wsf  = (float*)d_ws;
    float* xinv = wsf;
    float* winv = xinv + BB;
    float* gtv  = winv + CC;
    float* ctmv = gtv + BB;
    float* hsum = ctmv + BB;
    float* hcnt = hsum + BB;

    haaml_rownorm<<<BB + CC, 128, 0, stream>>>(x, w, xinv, winv);
    haaml_gt<<<BB, 128, 0, stream>>>(x, w, xinv, winv, label, gtv, ctmv, hsum, hcnt);

    dim3 grid(CC / 80, BB / 128);   // 625 x 32
    haaml_gemm<<<grid, 256, 0, stream>>>(x, w, xinv, winv, ctmv, label, out, hsum, hcnt);

    haaml_finalize<<<BB / 256, 256, 0, stream>>>(gtv, hsum, hcnt, label, out);

    const float* Hsrc = out + (size_t)BB * CC + 1 + 2 * BB;
    haaml_reg<<<1, 1024, 0, stream>>>(Hsrc, out + (size_t)BB * CC);
}